// MaxPlusDense_10393820857004
// MI455X (gfx1250) — compile-verified
//
#include <hip/hip_runtime.h>
#include <hip/hip_bf16.h>
#include <stdint.h>

// Tropical (max-plus) dense: out[b,u] = max(bias[u], max_i (x[b,i] - kernel[i,u]))
// B=256, D=1024, U=1024 fp32. WMMA cannot express the (max,+) semiring, so the
// CDNA5-specific paths used are async global->LDS copies (ASYNCcnt) for
// double-buffered tiling and fp32 global atomic max for a deterministic split-K.

#define BM 64
#define BN 64
#define BK 32
#define KSPLIT 4
#define LDX 68   // BM + 4 pad: conflict-free, keeps 16B alignment for b128 LDS reads
#define LDK 68   // BN + 4 pad
#define NTHREADS 256

struct __align__(16) F4 { float v[4]; };

#if defined(__gfx1250__) && \
    __has_builtin(__builtin_amdgcn_global_load_async_to_lds_b32) && \
    __has_builtin(__builtin_amdgcn_global_load_async_to_lds_b128) && \
    __has_builtin(__builtin_amdgcn_s_wait_asynccnt)
#define USE_ASYNC_LDS 1
#else
#define USE_ASYNC_LDS 0
#endif

#if USE_ASYNC_LDS
typedef int vec4i __attribute__((vector_size(4 * sizeof(int))));
typedef __attribute__((address_space(1))) int   as1_int_t;
typedef __attribute__((address_space(3))) int   as3_int_t;
typedef __attribute__((address_space(1))) vec4i as1_vec4i_t;
typedef __attribute__((address_space(3))) vec4i as3_vec4i_t;

__device__ __forceinline__ void tile_wait_async() {
  __builtin_amdgcn_s_wait_asynccnt(0);
}
__device__ __forceinline__ void copy_b32(const float* g, float* l) {
  __builtin_amdgcn_global_load_async_to_lds_b32((as1_int_t*)g, (as3_int_t*)l, 0, 0);
}
__device__ __forceinline__ void copy_b128(const float* g, float* l) {
  __builtin_amdgcn_global_load_async_to_lds_b128((as1_vec4i_t*)g, (as3_vec4i_t*)l, 0, 0);
}
#else
__device__ __forceinline__ void tile_wait_async() {}
__device__ __forceinline__ void copy_b32(const float* g, float* l) {
  *l = *g;
}
__device__ __forceinline__ void copy_b128(const float* g, float* l) {
  *(F4*)l = *(const F4*)g;
}
#endif

// out[b][u] = bias[u]  (implements the elementwise maximum(out, bias) term and
// re-initializes the poisoned output buffer every call -> deterministic replay)
__global__ __launch_bounds__(256) void mpd_init_kernel(float* __restrict__ out,
                                                       const float* __restrict__ bias,
                                                       int total, int U) {
  int idx = blockIdx.x * blockDim.x + threadIdx.x;
  if (idx < total) out[idx] = bias[idx % U];
}

__global__ __launch_bounds__(NTHREADS) void mpd_main_kernel(
    const float* __restrict__ x,     // [B, D] row-major
    const float* __restrict__ kern,  // [D, U] row-major
    float* __restrict__ out,         // [B, U]
    int D, int U, int kPerSplit) {
  // Double-buffered tiles. sX is k-major (transposed x) so 4 consecutive rows
  // load with one b128; +4 pad keeps banks clean and 16B alignment.
  __shared__ __align__(16) float sX[2][BK][LDX];
  __shared__ __align__(16) float sK[2][BK][LDK];

  const int tid  = threadIdx.x;
  const int lane = tid & 31;   // k_local for x staging
  const int wave = tid >> 5;   // 8 waves
  const int tx   = tid & 15;   // column group (4 cols each)
  const int ty   = tid >> 4;   // row group    (4 rows each)

  const int n0    = blockIdx.x * BN;
  const int m0    = blockIdx.y * BM;
  const int kbase = blockIdx.z * kPerSplit;
  const int numTiles = kPerSplit / BK;   // 8

  float acc[4][4];
#pragma unroll
  for (int i = 0; i < 4; ++i)
#pragma unroll
    for (int j = 0; j < 4; ++j) acc[i][j] = -__builtin_inff();

  auto loadTiles = [&](int buf, int kt) {
    const int k0 = kbase + kt * BK;
    // X tile, transposed scatter: sX[k_local][m] = x[m0+m][k0+k_local].
    // Per wave: lanes read 32 consecutive k (128B coalesced), 8 rows per wave.
    {
      const float* xg = x + (size_t)(m0 + 8 * wave) * D + k0 + lane;
      float* xl = &sX[buf][lane][8 * wave];
#pragma unroll
      for (int i = 0; i < 8; ++i) copy_b32(xg + (size_t)i * D, xl + i);
    }
    // K tile, contiguous: 2 x b128 per thread covers 32x64 floats.
    {
#pragma unroll
      for (int h = 0; h < 2; ++h) {
        const int f   = tid + h * NTHREADS;
        const int row = f >> 4;          // 0..31
        const int col = (f & 15) << 2;   // 0..60, step 4
        copy_b128(kern + (size_t)(k0 + row) * U + n0 + col, &sK[buf][row][col]);
      }
    }
  };

  loadTiles(0, 0);

  for (int kt = 0; kt < numTiles; ++kt) {
    tile_wait_async();     // this wave's async copies for tile kt are done
    __syncthreads();       // everyone's are done -> tile kt visible; buf^1 free
    if (kt + 1 < numTiles) loadTiles((kt + 1) & 1, kt + 1);  // overlap with compute

    const int buf = kt & 1;
    const float* sxp = &sX[buf][0][ty * 4];
    const float* skp = &sK[buf][0][tx * 4];

#pragma unroll 8
    for (int k = 0; k < BK; k += 2) {
      F4 a0 = *(const F4*)(sxp + (size_t)k * LDX);
      F4 a1 = *(const F4*)(sxp + (size_t)(k + 1) * LDX);
      F4 b0 = *(const F4*)(skp + (size_t)k * LDK);
      F4 b1 = *(const F4*)(skp + (size_t)(k + 1) * LDK);
#pragma unroll
      for (int i = 0; i < 4; ++i) {
#pragma unroll
        for (int j = 0; j < 4; ++j) {
          // fmax(fmax(d0,d1),acc) -> v_max3_f32 opportunity
          acc[i][j] = fmaxf(fmaxf(a0.v[i] - b0.v[j], a1.v[i] - b1.v[j]), acc[i][j]);
        }
      }
    }
    // No trailing barrier needed: the next iteration's wait+barrier precedes
    // any overwrite of the buffer just computed on.
  }

  // Split-K merge: fp32 max is exactly associative/commutative -> deterministic.
#pragma unroll
  for (int i = 0; i < 4; ++i) {
    float* orow = out + (size_t)(m0 + ty * 4 + i) * U + n0 + tx * 4;
#pragma unroll
    for (int j = 0; j < 4; ++j) {
      __hip_atomic_fetch_max(&orow[j], acc[i][j], __ATOMIC_RELAXED,
                             __HIP_MEMORY_SCOPE_AGENT);
    }
  }
}

extern "C" void kernel_launch(void* const* d_in, const int* in_sizes, int n_in,
                              void* d_out, int out_size, void* d_ws, size_t ws_size,
                              hipStream_t stream) {
  (void)n_in; (void)d_ws; (void)ws_size; (void)out_size;
  const float* x    = (const float*)d_in[0];  // [B, D]
  const float* kern = (const float*)d_in[1];  // [D, U]
  const float* bias = (const float*)d_in[2];  // [U]
  float* out = (float*)d_out;                 // [B, U]

  const int U = in_sizes[2];
  const int D = in_sizes[1] / U;
  const int B = in_sizes[0] / D;
  const int total = B * U;

  mpd_init_kernel<<<(total + 255) / 256, 256, 0, stream>>>(out, bias, total, U);

  dim3 grid(U / BN, B / BM, KSPLIT);   // 16 x 4 x 4 = 256 blocks
  mpd_main_kernel<<<grid, NTHREADS, 0, stream>>>(x, kern, out, D, U, D / KSPLIT);
}